// UnwrappingLoss_9861244912359
// MI455X (gfx1250) — compile-verified
//
#include <hip/hip_runtime.h>
#include <math.h>
#include <stdint.h>

// UnwrappingLoss on MI455X (gfx1250).
// d2 tile = A(16x4) * B(4x16) + C via V_WMMA_F32_16X16X4_F32 with augmented
// coords A=(-2x,-2y,-2z,1), B=(x,y,z,|q|^2), C[m][n]=|qm|^2.
// q (256KB augmented) is L2-resident; B-feed is staged to LDS in 32KB chunks
// by the Tensor Data Mover (tensor_load_to_lds + s_wait_tensorcnt), double
// buffered; the WMMA loop runs off software-pipelined ds_load_b64 (2 tiles
// in flight, unclamped prefetch: LDS OOB reads return 0 and dead values are
// discarded). Selection = per-lane sorted top-8 lists (sentinel eps^2),
// merged across lanes with shfl_xor bitonic merges. Deterministic
// reductions (no float atomics).

#define NPTS 16384
#define EPS2 0.25f                    // eps^2: d2 >= eps^2 contributes 0
#define WPB 4                         // waves per block
#define NBLOCKS (NPTS / (WPB * 16))   // 256 blocks
#define NWAVES (NPTS / 16)            // 1024 wave strips
#define CHUNK 2048                    // staged columns per chunk
#define NCHUNK (NPTS / CHUNK)         // 8
#define TPC (CHUNK / 16)              // 128 tiles per chunk
#define CHUNK_BYTES (CHUNK * 16)      // 32768

typedef __attribute__((ext_vector_type(2))) float v2f;
typedef __attribute__((ext_vector_type(8))) float v8f;
typedef __attribute__((ext_vector_type(4))) unsigned int v4u;
typedef __attribute__((ext_vector_type(8))) int v8i;
typedef __attribute__((ext_vector_type(4))) int v4i;

// Branch-guarded sorted insert of v into ascending 8-list (keep 8 smallest).
__device__ __forceinline__ void insert8(float (&L)[8], float v) {
    if (v < L[7]) {
        float n0 = fminf(L[0], v);
        float n1 = fminf(L[1], fmaxf(v, L[0]));
        float n2 = fminf(L[2], fmaxf(v, L[1]));
        float n3 = fminf(L[3], fmaxf(v, L[2]));
        float n4 = fminf(L[4], fmaxf(v, L[3]));
        float n5 = fminf(L[5], fmaxf(v, L[4]));
        float n6 = fminf(L[6], fmaxf(v, L[5]));
        float n7 = fminf(L[7], fmaxf(v, L[6]));
        L[0] = n0; L[1] = n1; L[2] = n2; L[3] = n3;
        L[4] = n4; L[5] = n5; L[6] = n6; L[7] = n7;
    }
}

// Tensor Data Mover: 1-D copy of CHUNK_BYTES from global to LDS.
// D# per CDNA5 ISA ch.8: data_size=8B, tensor_dim0=tile_dim0=4096,
// stride=4096, count=1, type=2 (image), workgroup_mask=0 (no cluster).
// 6-arg builtin form (clang-23 / therock headers).
__device__ __forceinline__ void tdm_issue(const void* gsrc, unsigned lds_off) {
    unsigned long long ga = (unsigned long long)(uintptr_t)gsrc;
    v4u g0;
    g0[0] = 1u;                                             // count=1, user D#
    g0[1] = lds_off;                                        // lds_addr
    g0[2] = (unsigned)(ga & 0xffffffffull);                 // global_addr lo
    g0[3] = (unsigned)((ga >> 32) & 0x01ffffffull) | 0x80000000u; // hi | type=2
    v8i g1;
    g1[0] = 0x00030000;       // [17:16] data_size=3 (8B); workgroup_mask=0
    g1[1] = 0x10000000;       // [63:48] tensor_dim0 = 4096 (low16)
    g1[2] = 0x00010000;       // [95:80] tensor_dim1 = 1
    g1[3] = 0x10000000;       // [127:112] tile_dim0 = 4096
    g1[4] = 0x00000001;       // [143:128] tile_dim1 = 1
    g1[5] = 4096;             // [191:160] tensor_dim0_stride = 4096
    g1[6] = 0;
    g1[7] = 0;
    v4i z4 = {0, 0, 0, 0};
    v8i z8 = {0, 0, 0, 0, 0, 0, 0, 0};
    __builtin_amdgcn_tensor_load_to_lds(g0, g1, z4, z4, z8, 0);
}

// Build augmented array qa[i] = (x, y, z, x^2+y^2+z^2).
__global__ __launch_bounds__(256) void prep_kernel(const float* __restrict__ q,
                                                   float4* __restrict__ qa) {
    int i = blockIdx.x * blockDim.x + threadIdx.x;
    if (i < NPTS) {
        float x = q[3 * i + 0];
        float y = q[3 * i + 1];
        float z = q[3 * i + 2];
        qa[i] = make_float4(x, y, z, x * x + y * y + z * z);
    }
}

// Generic range: depth-1 software pipeline. Prefetch is unclamped: reading one
// tile past t1 stays inside the 64KB stage allocation (or returns 0 if OOB)
// and the value is discarded at loop exit.
__device__ __forceinline__ void run_tiles(const char* sbase, int t0, int t1,
                                          v2f a, v8f cc, float (&lst)[8][8],
                                          int boff) {
    if (t0 >= t1) return;
    v2f b = *(const v2f*)(sbase + t0 * 256 + boff);
    for (int t = t0; t < t1; ++t) {
        v2f bn = *(const v2f*)(sbase + (t + 1) * 256 + boff);
        v8f d = __builtin_amdgcn_wmma_f32_16x16x4_f32(
            false, a, false, b, (short)0, cc, false, false);
#pragma unroll
        for (int r = 0; r < 8; ++r) insert8(lst[r], d[r]);
        b = bn;
    }
}

// Full chunk: 2 tiles/iteration -> two independent WMMAs back-to-back,
// prefetch distance 2 tiles (unclamped), two interleavable insert chains.
__device__ __forceinline__ void run_tiles2(const char* sbase,
                                           v2f a, v8f cc, float (&lst)[8][8],
                                           int boff) {
    v2f b0 = *(const v2f*)(sbase + 0 * 256 + boff);
    v2f b1 = *(const v2f*)(sbase + 1 * 256 + boff);
    for (int t = 0; t < TPC; t += 2) {
        v2f p0 = *(const v2f*)(sbase + (t + 2) * 256 + boff);
        v2f p1 = *(const v2f*)(sbase + (t + 3) * 256 + boff);
        v8f d0 = __builtin_amdgcn_wmma_f32_16x16x4_f32(
            false, a, false, b0, (short)0, cc, false, false);
        v8f d1 = __builtin_amdgcn_wmma_f32_16x16x4_f32(
            false, a, false, b1, (short)0, cc, false, false);
#pragma unroll
        for (int r = 0; r < 8; ++r) insert8(lst[r], d0[r]);
#pragma unroll
        for (int r = 0; r < 8; ++r) insert8(lst[r], d1[r]);
        b0 = p0;
        b1 = p1;
    }
}

// The one tile containing the self-distance: mask the diagonal element.
__device__ __forceinline__ void run_diag_tile(const char* sbase, int t,
                                              v2f a, v8f cc, float (&lst)[8][8],
                                              int boff, int l15, bool hi) {
    v2f b = *(const v2f*)(sbase + t * 256 + boff);
    v8f d = __builtin_amdgcn_wmma_f32_16x16x4_f32(
        false, a, false, b, (short)0, cc, false, false);
#pragma unroll
    for (int r = 0; r < 8; ++r) {
        float v = (l15 == r + (hi ? 8 : 0)) ? 1e9f : d[r];
        insert8(lst[r], v);
    }
}

__global__ __launch_bounds__(128) void knn_wmma_kernel(const float4* __restrict__ qa,
                                                       float* __restrict__ wsum) {
    __shared__ float4 stage4[2 * CHUNK];   // 2 x 32KB TDM double buffer (64KB)

    const int lane = threadIdx.x & 31;
    const int wv = threadIdx.x >> 5;
    // Force scalar (SGPR) strip id -> uniform control flow, no saveexec in loop.
    const int strip = __builtin_amdgcn_readfirstlane((int)(blockIdx.x * WPB + wv));
    const int base_m = strip << 4;
    const bool hi = lane >= 16;
    const int l15 = lane & 15;

    // ---- Loop-invariant A (16x4, 2 VGPRs): row m = (-2x,-2y,-2z,1).
    //   lanes 0-15: (K0,K1); lanes 16-31: (K2,K3)  [ISA 7.12.2]
    float4 qm = qa[base_m + l15];
    v2f a;
    a.x = hi ? (-2.0f * qm.z) : (-2.0f * qm.x);
    a.y = hi ? 1.0f           : (-2.0f * qm.y);

    // ---- Loop-invariant C: C[m][n] = |q_m|^2 ; VGPR r -> row r (lo) / r+8 (hi)
    v8f cc;
#pragma unroll
    for (int r = 0; r < 8; ++r)
        cc[r] = qa[base_m + r + (hi ? 8 : 0)].w;

    // ---- Per-lane top-8 lists per row slot, seeded with eps^2.
    float lst[8][8];
#pragma unroll
    for (int r = 0; r < 8; ++r)
#pragma unroll
        for (int k = 0; k < 8; ++k) lst[r][k] = EPS2;

    const int diag_chunk = strip >> 7;   // chunk containing this strip's diagonal
    const int diag_tl = strip & (TPC - 1);
    const int boff = (l15 << 4) + (hi ? 8 : 0);   // byte offset of B lane data
    const unsigned lds_base = (unsigned)(uintptr_t)stage4;
    const char* sb0 = (const char*)stage4;

    // Prime the pipeline: TDM chunk 0 into buffer 0.
    if (wv == 0) tdm_issue(qa, lds_base);

    for (int c = 0; c < NCHUNK; ++c) {
        if (wv == 0) __builtin_amdgcn_s_wait_tensorcnt(0);  // chunk c resident
        __syncthreads();                                    // publish to block
        if (wv == 0 && c + 1 < NCHUNK)                      // overlap next DMA
            tdm_issue(qa + (size_t)(c + 1) * CHUNK,
                      lds_base + (unsigned)(((c + 1) & 1) * CHUNK_BYTES));

        const char* s = sb0 + (c & 1) * CHUNK_BYTES;
        if (c == diag_chunk) {
            run_tiles(s, 0, diag_tl, a, cc, lst, boff);
            run_diag_tile(s, diag_tl, a, cc, lst, boff, l15, hi);
            run_tiles(s, diag_tl + 1, TPC, a, cc, lst, boff);
        } else {
            run_tiles2(s, a, cc, lst, boff);
        }
        __syncthreads();   // all waves done with buffer (c&1) before reuse
    }

    // ---- Cross-lane merge: 16-lane group shares one row per slot.
    //      4 rounds of shfl_xor sorted-merge (keep 8 smallest, bitonic clean).
#pragma unroll
    for (int r = 0; r < 8; ++r) {
#pragma unroll
        for (int m = 1; m <= 8; m <<= 1) {
            float P[8], T[8];
#pragma unroll
            for (int k = 0; k < 8; ++k) P[k] = __shfl_xor(lst[r][k], m, 32);
#pragma unroll
            for (int k = 0; k < 8; ++k) T[k] = fminf(lst[r][k], P[7 - k]);
#define CE(i, j) { float lo_ = fminf(T[i], T[j]); float hi_ = fmaxf(T[i], T[j]); T[i] = lo_; T[j] = hi_; }
            CE(0, 4) CE(1, 5) CE(2, 6) CE(3, 7)
            CE(0, 2) CE(1, 3) CE(4, 6) CE(5, 7)
            CE(0, 1) CE(2, 3) CE(4, 5) CE(6, 7)
#undef CE
#pragma unroll
            for (int k = 0; k < 8; ++k) lst[r][k] = T[k];
        }
    }

    // ---- Row losses: sum relu(eps - sqrt(max(d2,0))). Lanes 0-15 hold the
    //      8 low rows (replicated), lanes 16-31 the 8 high rows.
    float s = 0.0f;
#pragma unroll
    for (int r = 0; r < 8; ++r)
#pragma unroll
        for (int k = 0; k < 8; ++k) {
            float dn = sqrtf(fmaxf(lst[r][k], 0.0f));
            s += fmaxf(0.5f - dn, 0.0f);
        }
    float tot = __shfl(s, 0, 32) + __shfl(s, 16, 32);
    if (lane == 0) wsum[strip] = tot;
}

__global__ __launch_bounds__(256) void finalize_kernel(const float* __restrict__ wsum,
                                                       float* __restrict__ out) {
    __shared__ float red[256];
    int t = threadIdx.x;
    float s = wsum[t] + wsum[t + 256] + wsum[t + 512] + wsum[t + 768];
    red[t] = s;
    __syncthreads();
    for (int k = 128; k > 0; k >>= 1) {
        if (t < k) red[t] += red[t + k];
        __syncthreads();
    }
    if (t == 0) out[0] = red[0] * (1.0f / (float)NPTS);
}

extern "C" void kernel_launch(void* const* d_in, const int* in_sizes, int n_in,
                              void* d_out, int out_size, void* d_ws, size_t ws_size,
                              hipStream_t stream) {
    (void)in_sizes; (void)n_in; (void)out_size; (void)ws_size;

    const float* q = (const float*)d_in[0];
    float* out = (float*)d_out;

    // Workspace: qa[NPTS] float4 (256 KB), then 1024 per-wave partial sums.
    float4* qa = (float4*)d_ws;
    float* wsum = (float*)((char*)d_ws + (size_t)NPTS * sizeof(float4));

    prep_kernel<<<NPTS / 256, 256, 0, stream>>>(q, qa);
    knn_wmma_kernel<<<NBLOCKS, 128, 0, stream>>>(qa, wsum);
    finalize_kernel<<<1, 256, 0, stream>>>(wsum, out);
}